// MultiFeatureRNNModel_3848290697253
// MI455X (gfx1250) — compile-verified
//
#include <hip/hip_runtime.h>
#include <math.h>

typedef float v2f __attribute__((ext_vector_type(2)));
typedef float v8f __attribute__((ext_vector_type(8)));

constexpr int Tt = 512;
constexpr int Ff = 16;
constexpr int H1 = 5;
constexpr int H2 = 3;
constexpr int WAVES = 4;          // waves per block
constexpr int BPB   = WAVES * 32; // batch rows per block (1 per lane)

// gfx1250 hardware V_TANH_F32 trans op (co-executes with VALU/WMMA).
#if __has_builtin(__builtin_amdgcn_tanhf)
__device__ __forceinline__ float fast_tanh(float x) { return __builtin_amdgcn_tanhf(x); }
#elif __has_builtin(__builtin_amdgcn_tanh_f32)
__device__ __forceinline__ float fast_tanh(float x) { return __builtin_amdgcn_tanh_f32(x); }
#else
__device__ __forceinline__ float fast_tanh(float x) { return tanhf(x); }
#endif

// Fused 2-layer tanh-RNN scan. Each wave owns 32 batch rows; the x @ W_ih1^T
// projection per timestep is done with chained v_wmma_f32_16x16x4_f32 (two
// 16x16 C tiles per wave), results routed through a per-wave LDS tile so each
// lane then runs the tiny recurrence in registers. A-tile loads for t+1 are
// software-pipelined against the WMMA + recurrence of step t.
__global__ __launch_bounds__(128, 1) void rnn2_fused_wmma(
    const float* __restrict__ x,
    const float* __restrict__ W_ih1, const float* __restrict__ W_hh1,
    const float* __restrict__ b_ih1, const float* __restrict__ b_hh1,
    const float* __restrict__ W_ih2, const float* __restrict__ W_hh2,
    const float* __restrict__ b_ih2, const float* __restrict__ b_hh2,
    const float* __restrict__ W_fc,  const float* __restrict__ b_fc,
    float* __restrict__ out)
{
    // [wave][batch-in-wave][h] ; inner dim padded to 9 -> conflict-free gather
    __shared__ float xp[WAVES][32][9];

    const int tid  = threadIdx.x;
    const int lane = tid & 31;
    const int wave = tid >> 5;
    const int half = lane >> 4;   // which K-pair this lane supplies (f32 A/B layout)
    const int n    = lane & 15;   // output column (hidden unit) in the wmma tile
    const int base = blockIdx.x * BPB + wave * 32;

    // ---- B matrix = W_ih1^T, 4x16 f32 layout, one chunk per K-group of 4.
    // Lane layout: vgpr v holds B[k = j*4 + v + 2*half, n].  Constant all kernel.
    v2f bmat[4];
#pragma unroll
    for (int j = 0; j < 4; ++j) {
        const int k0 = j * 4 + 2 * half;
        float w0 = (n < H1) ? W_ih1[n * Ff + k0]     : 0.0f;
        float w1 = (n < H1) ? W_ih1[n * Ff + k0 + 1] : 0.0f;
        v2f b; b.x = w0; b.y = w1;
        bmat[j] = b;
    }
    // Fused layer-1 bias goes into the C initializer (column n is per-lane fixed).
    const float cinit = (n < H1) ? (b_ih1[n] + b_hh1[n]) : 0.0f;

    // ---- tiny recurrence weights -> registers (uniform addresses -> s_loads)
    float whh1[H1][H1], wih2[H2][H1], whh2[H2][H2], b2s[H2], wfc[H2];
#pragma unroll
    for (int i = 0; i < H1; ++i)
#pragma unroll
        for (int j = 0; j < H1; ++j) whh1[i][j] = W_hh1[i * H1 + j];
#pragma unroll
    for (int i = 0; i < H2; ++i) {
#pragma unroll
        for (int j = 0; j < H1; ++j) wih2[i][j] = W_ih2[i * H1 + j];
#pragma unroll
        for (int j = 0; j < H2; ++j) whh2[i][j] = W_hh2[i * H2 + j];
        b2s[i] = b_ih2[i] + b_hh2[i];
        wfc[i] = W_fc[i];
    }
    const float bfc = b_fc[0];

    // A-matrix row pointers: lane L is row m = L%16 of each 16-row tile,
    // supplying the contiguous float2 at k = j*4 + 2*half (f32 A layout).
    const float* rowA0  = x + (size_t)(base +      (lane & 15)) * (Tt * Ff) + 2 * half;
    const float* rowA1  = x + (size_t)(base + 16 + (lane & 15)) * (Tt * Ff) + 2 * half;
    const float* rowOwn = x + (size_t)(base + lane) * (Tt * Ff);

    float h1[H1], h2[H2];
#pragma unroll
    for (int i = 0; i < H1; ++i) h1[i] = 0.0f;
#pragma unroll
    for (int i = 0; i < H2; ++i) h2[i] = 0.0f;

    // ---- prologue: load A chunks for t = 0 (a[0..3]=tile0, a[4..7]=tile1)
    v2f a[8];
#pragma unroll
    for (int j = 0; j < 4; ++j) {
        a[j]     = __builtin_nontemporal_load((const v2f*)(rowA0 + j * 4));
        a[4 + j] = __builtin_nontemporal_load((const v2f*)(rowA1 + j * 4));
    }

    for (int t = 0; t < Tt; ++t) {
        // Prefetch the lane-owned row ahead of the pipeline (global_prefetch_b8).
        const int tp = (t + 8 < Tt) ? (t + 8) : (Tt - 1);
        __builtin_prefetch(rowOwn + (size_t)tp * Ff, 0, 1);

        // ---- issue next timestep's 8 A-chunk loads (waits land next iteration)
        v2f anx[8];
        {
            const int tn = (t + 1 < Tt) ? (t + 1) : t;  // clamped; last iter redundant
            const float* p0 = rowA0 + (size_t)tn * Ff;
            const float* p1 = rowA1 + (size_t)tn * Ff;
#pragma unroll
            for (int j = 0; j < 4; ++j) {
                anx[j]     = __builtin_nontemporal_load((const v2f*)(p0 + j * 4));
                anx[4 + j] = __builtin_nontemporal_load((const v2f*)(p1 + j * 4));
            }
        }

        // ---- projection tiles: C[m,n] = b1[n] + sum_k x[row, t, k] * W_ih1[n,k]
        v8f c0, c1;
#pragma unroll
        for (int i = 0; i < 8; ++i) { c0[i] = cinit; c1[i] = cinit; }
#pragma unroll
        for (int j = 0; j < 4; ++j) {
            c0 = __builtin_amdgcn_wmma_f32_16x16x4_f32(
                false, a[j], false, bmat[j], (short)0, c0, false, false);
            c1 = __builtin_amdgcn_wmma_f32_16x16x4_f32(
                false, a[4 + j], false, bmat[j], (short)0, c1, false, false);
        }

        // ---- scatter C tiles to LDS: vgpr r of lanes 0-15 is M=r, of lanes 16-31 is M=8+r
        if (n < H1) {
#pragma unroll
            for (int r = 0; r < 8; ++r) xp[wave][half * 8 + r][n]      = c0[r];
#pragma unroll
            for (int r = 0; r < 8; ++r) xp[wave][16 + half * 8 + r][n] = c1[r];
        }
        // No block barrier needed: the scatter->gather hazard is same-wave only
        // (each wave owns its LDS slice) and same-wave LDS ops execute in order
        // (ISA 7.3, DScnt); the compiler inserts s_wait_dscnt for the results.

        float xv[H1];
#pragma unroll
        for (int i = 0; i < H1; ++i) xv[i] = xp[wave][lane][i];

        // ---- per-lane recurrence (lane owns batch base+lane)
        float h1n[H1];
#pragma unroll
        for (int i = 0; i < H1; ++i) {
            float s = xv[i];
#pragma unroll
            for (int j = 0; j < H1; ++j) s += whh1[i][j] * h1[j];
            h1n[i] = fast_tanh(s);
        }
#pragma unroll
        for (int i = 0; i < H1; ++i) h1[i] = h1n[i];

        float h2n[H2];
#pragma unroll
        for (int i = 0; i < H2; ++i) {
            float s = b2s[i];
#pragma unroll
            for (int j = 0; j < H1; ++j) s += wih2[i][j] * h1n[j];
#pragma unroll
            for (int j = 0; j < H2; ++j) s += whh2[i][j] * h2[j];
            h2n[i] = fast_tanh(s);
        }
#pragma unroll
        for (int i = 0; i < H2; ++i) h2[i] = h2n[i];

        // rotate the pipeline
#pragma unroll
        for (int j = 0; j < 8; ++j) a[j] = anx[j];
    }

    float y = bfc;
#pragma unroll
    for (int j = 0; j < H2; ++j) y += wfc[j] * h2[j];
#if __has_builtin(__builtin_amdgcn_rcpf)
    out[base + lane] = __builtin_amdgcn_rcpf(1.0f + __expf(-y));
#else
    out[base + lane] = 1.0f / (1.0f + __expf(-y));
#endif
}

extern "C" void kernel_launch(void* const* d_in, const int* in_sizes, int n_in,
                              void* d_out, int out_size, void* d_ws, size_t ws_size,
                              hipStream_t stream) {
    const float* x     = (const float*)d_in[0];
    const float* W_ih1 = (const float*)d_in[1];
    const float* W_hh1 = (const float*)d_in[2];
    const float* b_ih1 = (const float*)d_in[3];
    const float* b_hh1 = (const float*)d_in[4];
    const float* W_ih2 = (const float*)d_in[5];
    const float* W_hh2 = (const float*)d_in[6];
    const float* b_ih2 = (const float*)d_in[7];
    const float* b_hh2 = (const float*)d_in[8];
    const float* W_fc  = (const float*)d_in[9];
    const float* b_fc  = (const float*)d_in[10];
    float* out = (float*)d_out;

    const int Btot = in_sizes[0] / (Tt * Ff);  // 8192
    const int grid = Btot / BPB;               // 64 workgroups of 4 waves

    rnn2_fused_wmma<<<grid, WAVES * 32, 0, stream>>>(
        x, W_ih1, W_hh1, b_ih1, b_hh1, W_ih2, W_hh2, b_ih2, b_hh2, W_fc, b_fc, out);
}